// GCN_2336462209053
// MI455X (gfx1250) — compile-verified
//
#include <hip/hip_runtime.h>

typedef __attribute__((ext_vector_type(2))) float v2f;
typedef __attribute__((ext_vector_type(8))) float v8f;
typedef int v4i_vs __attribute__((vector_size(16)));  // matches builtin param

// ---- gfx1250 async global->LDS staging (guarded; falls back to float4 copy)
#define AS1 __attribute__((address_space(1)))
#define AS3 __attribute__((address_space(3)))

#if defined(__has_builtin)
#if __has_builtin(__builtin_amdgcn_global_load_async_to_lds_b128) && \
    __has_builtin(__builtin_amdgcn_s_wait_asynccnt)
#define HAVE_ASYNC_LDS 1
#endif
#endif
#ifndef HAVE_ASYNC_LDS
#define HAVE_ASYNC_LDS 0
#endif

__device__ __forceinline__ void stage16(float* lds, const float* g) {
#if HAVE_ASYNC_LDS
    __builtin_amdgcn_global_load_async_to_lds_b128(
        (AS1 v4i_vs*)g, (AS3 v4i_vs*)lds, 0, 0);
#else
    *(float4*)lds = *(const float4*)g;
#endif
}

__device__ __forceinline__ void stage_wait() {
#if HAVE_ASYNC_LDS
    __builtin_amdgcn_s_wait_asynccnt(0);
#endif
}

// ---------------------------------------------------------------------------
// gcn_norm pieces
// ---------------------------------------------------------------------------
__global__ void init_deg_kernel(float* __restrict__ deg, int n) {
    int i = blockIdx.x * blockDim.x + threadIdx.x;
    if (i < n) deg[i] = 1.0f;  // self-loop contribution (fill_value = 1.0)
}

__global__ void accum_deg_kernel(const long long* __restrict__ ei,
                                 const float* __restrict__ ew,
                                 float* __restrict__ deg, int n_edges) {
    int e = blockIdx.x * blockDim.x + threadIdx.x;
    if (e < n_edges) {
        int col = (int)ei[(size_t)n_edges + e];  // target row of edge_index[1]
        atomicAdd(&deg[col], ew[e]);
    }
}

__global__ void dis_kernel(float* __restrict__ deg, int n) {
    int i = blockIdx.x * blockDim.x + threadIdx.x;
    if (i < n) {
        float d = deg[i];
        deg[i] = (d > 0.0f) ? rsqrtf(d) : 0.0f;  // deg^-1/2, inf -> 0
    }
}

__global__ void norm_kernel(const long long* __restrict__ ei,
                            const float* __restrict__ ew,
                            const float* __restrict__ dis,
                            float* __restrict__ norm, int n_edges) {
    int e = blockIdx.x * blockDim.x + threadIdx.x;
    if (e < n_edges) {
        int r = (int)ei[e];
        int c = (int)ei[(size_t)n_edges + e];
        norm[e] = dis[r] * ew[e] * dis[c];
    }
}

// ---------------------------------------------------------------------------
// Dense GEMM: out[M,NC] = A[M,K] @ W[K,NC] using V_WMMA_F32_16X16X4_F32.
// One block = one 16-row m-tile; wave w computes 16x16 n-tile w.
// A tile and W staged in LDS via async direct-to-LDS loads; A rows padded by
// 4 floats (keeps 16B alignment for B128 and distinct banks for the 16-lane
// strided fragment reads: stride mod 64 = 4 dwords).
// ---------------------------------------------------------------------------
template <int K, int NC>
__global__ __launch_bounds__(NC * 2) void gemm_wmma_kernel(
    const float* __restrict__ A, const float* __restrict__ W,
    float* __restrict__ out, int M) {
    constexpr int PAD_K = K + 4;
    __shared__ float Alds[16 * PAD_K];
    __shared__ float Wlds[K * NC];

    const int tid = threadIdx.x;
    const int lane = tid & 31;
    const int wave = tid >> 5;
    const int m0 = blockIdx.x * 16;

    // Stage A tile (16 x K): coalesced 16B async loads straight into LDS.
    for (int idx = tid; idx < 16 * (K / 4); idx += NC * 2) {
        int r = idx / (K / 4);
        int c4 = (idx % (K / 4)) * 4;
        int gr = m0 + r;
        float* dst = &Alds[r * PAD_K + c4];
        if (gr < M) {
            stage16(dst, A + (size_t)gr * K + c4);
        } else {
            dst[0] = 0.f; dst[1] = 0.f; dst[2] = 0.f; dst[3] = 0.f;
        }
    }
    // Stage W (K x NC), contiguous 16B async loads.
    for (int idx = tid; idx < (K * NC) / 4; idx += NC * 2) {
        stage16(&Wlds[idx * 4], W + (size_t)idx * 4);
    }
    stage_wait();
    __syncthreads();

    const int mn = lane & 15;   // A-frag row / B-frag col
    const int kh = lane >> 4;   // half-wave selects K pair
    const int n0 = wave * 16;

    v8f acc = {};
#pragma unroll
    for (int kk = 0; kk < K; kk += 4) {
        v2f a, b;
        const float* ap = &Alds[mn * PAD_K + kk + 2 * kh];
        a.x = ap[0];
        a.y = ap[1];
        b.x = Wlds[(kk + 2 * kh) * NC + n0 + mn];
        b.y = Wlds[(kk + 2 * kh + 1) * NC + n0 + mn];
        acc = __builtin_amdgcn_wmma_f32_16x16x4_f32(
            false, a, false, b, (short)0, acc, false, false);
    }

#pragma unroll
    for (int r = 0; r < 8; ++r) {
        int row = m0 + r + 8 * kh;  // C/D layout: lanes16-31 hold M = r+8
        if (row < M) out[(size_t)row * NC + n0 + mn] = acc[r];
    }
}

// ---------------------------------------------------------------------------
// Aggregation: aggr = selfnorm * t   (self-loop term, also zero-init),
// then atomic scatter over edges, then bias (+ ReLU).
// ---------------------------------------------------------------------------
template <int C>
__global__ void init_self_kernel(const float* __restrict__ t,
                                 const float* __restrict__ dis,
                                 float* __restrict__ aggr, int n) {
    int idx = blockIdx.x * blockDim.x + threadIdx.x;
    int total = n * (C / 4);
    if (idx >= total) return;
    int i = idx / (C / 4);
    int c4 = (idx % (C / 4)) * 4;
    float d = dis[i];
    float s = d * d;  // dis[i] * 1.0 * dis[i]
    const float4 v = *(const float4*)(t + (size_t)i * C + c4);
    float4 o = make_float4(s * v.x, s * v.y, s * v.z, s * v.w);
    *(float4*)(aggr + (size_t)i * C + c4) = o;
}

template <int C>
__global__ void scatter_kernel(const float* __restrict__ t,
                               const float* __restrict__ norm,
                               const long long* __restrict__ ei,
                               float* __restrict__ aggr, int n_edges) {
    int idx = blockIdx.x * blockDim.x + threadIdx.x;
    int total = n_edges * (C / 4);
    if (idx >= total) return;
    int e = idx / (C / 4);
    int c4 = (idx % (C / 4)) * 4;
    int r = (int)ei[e];                        // source
    int c = (int)ei[(size_t)n_edges + e];      // target
    float w = norm[e];
    const float4 v = *(const float4*)(t + (size_t)r * C + c4);
    float* dst = aggr + (size_t)c * C + c4;
    atomicAdd(dst + 0, w * v.x);
    atomicAdd(dst + 1, w * v.y);
    atomicAdd(dst + 2, w * v.z);
    atomicAdd(dst + 3, w * v.w);
}

template <int C, bool RELU>
__global__ void bias_act_kernel(const float* __restrict__ aggr,
                                const float* __restrict__ b,
                                float* __restrict__ out, int n) {
    int idx = blockIdx.x * blockDim.x + threadIdx.x;
    if (idx >= n * C) return;
    int c = idx % C;
    float v = aggr[idx] + b[c];
    out[idx] = RELU ? fmaxf(v, 0.0f) : v;
}

// ---------------------------------------------------------------------------
static inline int cdiv(int a, int b) { return (a + b - 1) / b; }

extern "C" void kernel_launch(void* const* d_in, const int* in_sizes, int n_in,
                              void* d_out, int out_size, void* d_ws,
                              size_t ws_size, hipStream_t stream) {
    const float* x      = (const float*)d_in[0];
    const long long* ei = (const long long*)d_in[1];  // int64 [2, E]
    const float* ew     = (const float*)d_in[2];
    const float* W1     = (const float*)d_in[3];
    const float* b1     = (const float*)d_in[4];
    const float* W2     = (const float*)d_in[5];
    const float* b2     = (const float*)d_in[6];
    const float* W3     = (const float*)d_in[7];
    const float* b3     = (const float*)d_in[8];
    float* out = (float*)d_out;

    const int IN_C = 128, HID = 64;
    const int E = in_sizes[2];
    const int N = in_sizes[0] / IN_C;

    // Workspace layout (floats): dis[N] | norm[E] | tA[N*64] | tB[N*64] | aggr[N*64]
    float* ws = (float*)d_ws;
    size_t off = 0;
    float* dis = ws + off;  off += ((size_t)N + 255) & ~(size_t)255;
    float* nrm = ws + off;  off += ((size_t)E + 255) & ~(size_t)255;
    float* tA  = ws + off;  off += (size_t)N * HID;
    float* tB  = ws + off;  off += (size_t)N * HID;
    float* agg = ws + off;

    const int mtiles = cdiv(N, 16);

    // ---- gcn_norm ----
    init_deg_kernel<<<cdiv(N, 256), 256, 0, stream>>>(dis, N);
    accum_deg_kernel<<<cdiv(E, 256), 256, 0, stream>>>(ei, ew, dis, E);
    dis_kernel<<<cdiv(N, 256), 256, 0, stream>>>(dis, N);
    norm_kernel<<<cdiv(E, 256), 256, 0, stream>>>(ei, ew, dis, nrm, E);

    // ---- layer 1: relu(agg(x @ W1) + b1) ----
    gemm_wmma_kernel<128, 64><<<mtiles, 128, 0, stream>>>(x, W1, tA, N);
    init_self_kernel<64><<<cdiv(N * 16, 256), 256, 0, stream>>>(tA, dis, agg, N);
    scatter_kernel<64><<<cdiv(E * 16, 256), 256, 0, stream>>>(tA, nrm, ei, agg, E);
    bias_act_kernel<64, true><<<cdiv(N * 64, 256), 256, 0, stream>>>(agg, b1, tB, N);

    // ---- layer 2: relu(agg(h1 @ W2) + b2) ----
    gemm_wmma_kernel<64, 64><<<mtiles, 128, 0, stream>>>(tB, W2, tA, N);
    init_self_kernel<64><<<cdiv(N * 16, 256), 256, 0, stream>>>(tA, dis, agg, N);
    scatter_kernel<64><<<cdiv(E * 16, 256), 256, 0, stream>>>(tA, nrm, ei, agg, E);
    bias_act_kernel<64, true><<<cdiv(N * 64, 256), 256, 0, stream>>>(agg, b2, tB, N);

    // ---- layer 3: agg(h2 @ W3) + b3 ----
    gemm_wmma_kernel<64, 32><<<mtiles, 64, 0, stream>>>(tB, W3, tA, N);
    init_self_kernel<32><<<cdiv(N * 8, 256), 256, 0, stream>>>(tA, dis, agg, N);
    scatter_kernel<32><<<cdiv(E * 8, 256), 256, 0, stream>>>(tA, nrm, ei, agg, E);
    bias_act_kernel<32, false><<<cdiv(N * 32, 256), 256, 0, stream>>>(agg, b3, out, N);
}